// ETE_ETV_Net_58712202936346
// MI455X (gfx1250) — compile-verified
//
#include <hip/hip_runtime.h>
#include <hip/hip_bf16.h>

typedef float v2f __attribute__((ext_vector_type(2)));
typedef float v8f __attribute__((ext_vector_type(8)));

#define NB   128        // batch
#define NN   256        // n
#define NSQ  65536      // n*n
#define NC   768        // 3n columns of w1
#define KSPLIT 32       // k-chunks over the square region (8 rows each)

// ---------------------------------------------------------------------------
// Kernel 1: per-batch ETE/ETV closed-form pipeline -> x3[b, 0..255]
// ---------------------------------------------------------------------------
__device__ __forceinline__ float block_sum256(float v, float* red) {
    const int t = threadIdx.x;
    red[t] = v;
    __syncthreads();
    #pragma unroll
    for (int s = 128; s > 0; s >>= 1) {
        if (t < s) red[t] += red[t + s];
        __syncthreads();
    }
    float r = red[0];
    __syncthreads();
    return r;
}

__global__ void prep_x_kernel(const float* __restrict__ in, float* __restrict__ x3) {
    __shared__ float red[256];
    const int b = blockIdx.x;
    const int i = threadIdx.x;            // row 0..255
    const float* row = in + (size_t)b * NSQ + (size_t)i * NN;

    // row sum under lower-triangular mask: sum_{j<=i}
    float rs = 0.0f;
    for (int j = 0; j <= i; ++j) rs += row[j];
    const float e0 = row[0];

    const float S0 = block_sum256(e0, red);        // column-0 sum
    const float x1 = rs + S0 - 2.0f * e0;          // etv #1
    const float T  = block_sum256(x1, red);
    const float x2 = (T - x1) * x1;                // ete on (n,1)
    const float S2 = block_sum256(x2, red);
    const float v  = (float)(i - 1) * x2 + S2;     // etv #2 after *tri
    x3[(size_t)b * NN + i] = v;
}

// ---------------------------------------------------------------------------
// Kernel 2: h1 partials = f @ w1 with y==z weight folding, fp32 WMMA 16x16x4.
// grid (48 n-tiles, 32 k-chunks), 256 threads = 8 waves, 1 C-tile per wave.
// Branch-free inner loops: chunk c owns input rows [8c, 8c+8); chunk 0 also
// handles the 256-wide x-part. Every 4-wide K step stays inside one row.
// ---------------------------------------------------------------------------
#define WMMA_F32(a, b, c) \
    __builtin_amdgcn_wmma_f32_16x16x4_f32(false, (a), false, (b), (short)0, (c), false, false)

__global__ void gemm1_kernel(const float* __restrict__ in,
                             const float* __restrict__ x3,
                             const float* __restrict__ w1,
                             float* __restrict__ h1part) {
    const int lane = threadIdx.x & 31;
    const int wave = threadIdx.x >> 5;          // M-tile 0..7
    const int half = lane >> 4;                 // 0/1 -> K offset 0/2
    const int l16  = lane & 15;
    const int brow = wave * 16 + l16;           // A row (batch)
    const int ncol = blockIdx.x * 16 + l16;     // B col (w1 column)

    v8f acc = {};

    // ---- x-part (folded K 0..255), chunk 0 only; uniform, branch-free ----
    if (blockIdx.y == 0) {
        const float* __restrict__ xr = x3 + (size_t)brow * NN;
        #pragma unroll 4
        for (int k = 0; k < NN; k += 4) {
            const int kk = k + 2 * half;
            v2f a, bf;
            a.x  = xr[kk];
            a.y  = xr[kk + 1];
            bf.x = __builtin_nontemporal_load(&w1[(size_t)kk * 513 * NC + ncol]);
            bf.y = __builtin_nontemporal_load(&w1[(size_t)(kk + 1) * 513 * NC + ncol]);
            acc = WMMA_F32(a, bf, acc);
        }
    }

    // ---- square region: 8 input rows per chunk, fully affine addressing ----
    const int i0 = blockIdx.y * 8;
    const float* __restrict__ inb = in + (size_t)brow * NSQ + (size_t)i0 * NN;
    const float* __restrict__ wyb = w1 + ((size_t)i0 * 513 + 1) * NC + ncol;

    for (int ii = 0; ii < 8; ++ii) {
        const float* __restrict__ inrow = inb + (size_t)ii * NN;
        const float* __restrict__ wy    = wyb + (size_t)ii * 513 * NC;
        #pragma unroll 4
        for (int t = 0; t < NN; t += 4) {
            const int tt = t + 2 * half;
            const float a0 = inrow[tt];
            const float a1 = inrow[tt + 1];
            v2f a, bf;
            a.x = a0 * a0;
            a.y = a1 * a1;
            // fold y and z weight rows (z row = y row + 256 rows)
            bf.x = __builtin_nontemporal_load(&wy[(size_t)tt * NC]) +
                   __builtin_nontemporal_load(&wy[(size_t)tt * NC + (size_t)256 * NC]);
            bf.y = __builtin_nontemporal_load(&wy[(size_t)(tt + 1) * NC]) +
                   __builtin_nontemporal_load(&wy[(size_t)(tt + 1) * NC + (size_t)256 * NC]);
            acc = WMMA_F32(a, bf, acc);
        }
    }

    // C/D layout: VGPR r -> M = r + 8*half, N = lane&15
    float* dst = h1part + (size_t)blockIdx.y * NB * NC;
    #pragma unroll
    for (int r = 0; r < 8; ++r) {
        const int bg = wave * 16 + r + 8 * half;
        dst[(size_t)bg * NC + ncol] = acc[r];
    }
}

// ---------------------------------------------------------------------------
// Kernel 3: deterministic k-split reduction + bias + relu + tiny MLP tail
// ---------------------------------------------------------------------------
__global__ void tail_kernel(const float* __restrict__ h1part,
                            const float* __restrict__ b1,
                            const float* __restrict__ w2,
                            const float* __restrict__ b2,
                            const float* __restrict__ w3,
                            const float* __restrict__ b3,
                            float* __restrict__ out) {
    __shared__ float h1[NC];
    __shared__ float h2[10];
    const int b = blockIdx.x;
    const int t = threadIdx.x;   // 256 threads

    for (int c = t; c < NC; c += 256) {
        float s = b1[c];
        for (int ch = 0; ch < KSPLIT; ++ch)   // fixed order -> deterministic
            s += h1part[((size_t)ch * NB + b) * NC + c];
        h1[c] = fmaxf(s, 0.0f);
    }
    __syncthreads();

    if (t < 10) {
        float s = b2[t];
        for (int c = 0; c < NC; ++c) s += h1[c] * w2[c * 10 + t];
        h2[t] = fmaxf(s, 0.0f);
    }
    __syncthreads();

    if (t < 10) {
        float s = b3[t];
        #pragma unroll
        for (int c2 = 0; c2 < 10; ++c2) s += h2[c2] * w3[c2 * 10 + t];
        out[(size_t)b * 10 + t] = s;
    }
}

// ---------------------------------------------------------------------------
extern "C" void kernel_launch(void* const* d_in, const int* in_sizes, int n_in,
                              void* d_out, int out_size, void* d_ws, size_t ws_size,
                              hipStream_t stream) {
    const float* in = (const float*)d_in[0];
    const float* w1 = (const float*)d_in[1];
    const float* b1 = (const float*)d_in[2];
    const float* w2 = (const float*)d_in[3];
    const float* b2 = (const float*)d_in[4];
    const float* w3 = (const float*)d_in[5];
    const float* b3 = (const float*)d_in[6];
    float* out = (float*)d_out;

    float* ws     = (float*)d_ws;
    float* x3     = ws;                    // 128*256 f32 = 128 KB
    float* h1part = ws + NB * NN;          // 32*128*768 f32 = 12.6 MB

    prep_x_kernel<<<NB, 256, 0, stream>>>(in, x3);
    gemm1_kernel<<<dim3(NC / 16, KSPLIT), 256, 0, stream>>>(in, x3, w1, h1part);
    tail_kernel<<<NB, 256, 0, stream>>>(h1part, b1, w2, b2, w3, b3, out);
}